// GCNEncoder_45140106281283
// MI455X (gfx1250) — compile-verified
//
#include <hip/hip_runtime.h>
#include <hip/hip_bf16.h>

typedef float v2f __attribute__((ext_vector_type(2)));
typedef float v8f __attribute__((ext_vector_type(8)));

#define D 128          // feature dim (fixed by the reference)
#define TILES_N 8      // 128/16 column tiles

// ---------------- zero fill ----------------
__global__ void zero_kernel(float* __restrict__ p, long long n) {
    long long i = (long long)blockIdx.x * blockDim.x + threadIdx.x;
    if (i < n) p[i] = 0.0f;
}

// ---------------- degree via atomics ----------------
__global__ void deg_kernel(const long long* __restrict__ dst, float* __restrict__ deg,
                           long long E) {
    long long e = (long long)blockIdx.x * blockDim.x + threadIdx.x;
    if (e < E) atomicAdd(&deg[(int)dst[e]], 1.0f);
}

// dinv[i] = rsqrt(deg[i] + 1)   (+1 = self loop; always > 0)
__global__ void dinv_kernel(float* __restrict__ deg, int n) {
    int i = blockIdx.x * blockDim.x + threadIdx.x;
    if (i < n) deg[i] = rsqrtf(deg[i] + 1.0f);
}

// ---------------- GEMM: T[N x 128] = A[N x 128] @ W[128 x 128] ----------------
// One wave per 16-row stripe; wave computes all 8 16x16 column tiles so A is
// read exactly once. Uses V_WMMA_F32_16X16X4_F32 (fp32 matrix path, wave32).
// A layout (16x4 f32): lanes 0-15 hold K=k,k+1 (v0,v1) for M=lane;
//                      lanes 16-31 hold K=k+2,k+3 for M=lane-16.
// B layout (4x16 f32): v0/v1 hold rows K=k(+2*half), K=k+1(+2*half), N=lane&15.
// C/D layout (16x16 f32): VGPR g -> M = g + 8*half, N = lane&15.
__global__ __launch_bounds__(256)
void gemm_wmma_kernel(const float* __restrict__ A, const float* __restrict__ W,
                      float* __restrict__ T, int N) {
    const int wave = threadIdx.x >> 5;
    const int lane = threadIdx.x & 31;
    const int rowTile = blockIdx.x * 8 + wave;
    const int m0 = rowTile * 16;
    if (m0 >= N) return;                 // uniform per wave -> EXEC stays all-1s
    const int half = lane >> 4;          // 0: lanes 0-15, 1: lanes 16-31
    const int l = lane & 15;

    v8f acc[TILES_N];
    const v8f vzero = {0.f, 0.f, 0.f, 0.f, 0.f, 0.f, 0.f, 0.f};
#pragma unroll
    for (int t = 0; t < TILES_N; ++t) acc[t] = vzero;

    const float* __restrict__ arow = A + (long long)(m0 + l) * D;

    for (int k = 0; k < D; k += 4) {
        v2f a;
        a.x = arow[k + 2 * half];
        a.y = arow[k + 2 * half + 1];
#pragma unroll
        for (int t = 0; t < TILES_N; ++t) {
            const int n0 = t * 16;
            v2f b;
            b.x = W[(k + 2 * half) * D + n0 + l];
            b.y = W[(k + 2 * half + 1) * D + n0 + l];
            acc[t] = __builtin_amdgcn_wmma_f32_16x16x4_f32(
                /*neg_a=*/false, a, /*neg_b=*/false, b,
                /*c_mod=*/(short)0, acc[t], /*reuse_a=*/false, /*reuse_b=*/false);
        }
    }

#pragma unroll
    for (int t = 0; t < TILES_N; ++t) {
        const int n0 = t * 16;
#pragma unroll
        for (int g = 0; g < 8; ++g) {
            const int row = m0 + g + half * 8;
            T[(long long)row * D + n0 + l] = acc[t][g];
        }
    }
}

// ---------------- scatter-aggregate: one wave per edge ----------------
// acc[dst] += T[src] * dinv[src]*dinv[dst]; 32 lanes x float4 = 128 features.
__global__ __launch_bounds__(256)
void scatter_kernel(const float* __restrict__ T,
                    const long long* __restrict__ src,
                    const long long* __restrict__ dst,
                    const float* __restrict__ dinv,
                    float* __restrict__ acc, long long E) {
    long long e = (long long)blockIdx.x * 8 + (threadIdx.x >> 5);
    if (e >= E) return;
    const int lane = threadIdx.x & 31;
    const int s = (int)src[e];
    const int d = (int)dst[e];
    const float norm = dinv[s] * dinv[d];
    const float4 v = ((const float4*)(T + (long long)s * D))[lane];
    float* out = acc + (long long)d * D + lane * 4;
    atomicAdd(out + 0, v.x * norm);
    atomicAdd(out + 1, v.y * norm);
    atomicAdd(out + 2, v.z * norm);
    atomicAdd(out + 3, v.w * norm);
}

// ---------------- finalize: add self-loop term + bias, ReLU ----------------
__global__ void finalize_kernel(const float* __restrict__ acc,
                                const float* __restrict__ T,
                                const float* __restrict__ dinv,
                                const float* __restrict__ bias,
                                float* __restrict__ out, int N) {
    long long idx = (long long)blockIdx.x * blockDim.x + threadIdx.x;
    if (idx >= (long long)N * D) return;
    const int row = (int)(idx >> 7);
    const int f = (int)(idx & (D - 1));
    const float di = dinv[row];
    const float v = acc[idx] + T[idx] * di * di + bias[f];
    out[idx] = v > 0.0f ? v : 0.0f;
}

extern "C" void kernel_launch(void* const* d_in, const int* in_sizes, int n_in,
                              void* d_out, int out_size, void* d_ws, size_t ws_size,
                              hipStream_t stream) {
    const float*     x   = (const float*)d_in[0];
    const long long* ei  = (const long long*)d_in[1];   // int64 [2, E]
    const float*     W1  = (const float*)d_in[2];
    const float*     b1  = (const float*)d_in[3];
    const float*     W2  = (const float*)d_in[4];
    const float*     b2  = (const float*)d_in[5];
    float*           out = (float*)d_out;

    const int       N = in_sizes[0] / D;          // 100000
    const long long E = (long long)in_sizes[1] / 2; // 1600000

    const long long* src = ei;
    const long long* dst = ei + E;

    // workspace layout: dinv[N] | T[N*D]
    float* dinv = (float*)d_ws;
    float* T    = dinv + N;

    const long long ND = (long long)N * D;
    const int thr = 256;
    const int blkN  = (N + thr - 1) / thr;
    const int blkND = (int)((ND + thr - 1) / thr);
    const int blkE  = (int)((E + thr - 1) / thr);
    const int blkEw = (int)((E + 7) / 8);          // wave-per-edge (8 waves/block)
    const int blkG  = (N / 16 + 7) / 8;            // gemm: 8 waves/block, 16 rows/wave

    // --- graph normalization (shared by both layers) ---
    zero_kernel<<<blkN, thr, 0, stream>>>(dinv, N);
    deg_kernel<<<blkE, thr, 0, stream>>>(dst, dinv, E);
    dinv_kernel<<<blkN, thr, 0, stream>>>(dinv, N);

    // --- layer 1: T = x@W1; out = relu(scatter(T) + T*dinv^2 + b1) ---
    gemm_wmma_kernel<<<blkG, thr, 0, stream>>>(x, W1, T, N);
    zero_kernel<<<blkND, thr, 0, stream>>>(out, ND);
    scatter_kernel<<<blkEw, thr, 0, stream>>>(T, src, dst, dinv, out, E);
    finalize_kernel<<<blkND, thr, 0, stream>>>(out, T, dinv, b1, out, N);

    // --- layer 2: T = out@W2; out = relu(scatter(T) + T*dinv^2 + b2) ---
    // (gemm reads `out` fully before the zero_kernel below; stream-ordered)
    gemm_wmma_kernel<<<blkG, thr, 0, stream>>>(out, W2, T, N);
    zero_kernel<<<blkND, thr, 0, stream>>>(out, ND);
    scatter_kernel<<<blkEw, thr, 0, stream>>>(T, src, dst, dinv, out, E);
    finalize_kernel<<<blkND, thr, 0, stream>>>(out, T, dinv, b2, out, N);
}